// HyperedgeToHyperedgeLayer_20220706030438
// MI455X (gfx1250) — compile-verified
//
#include <hip/hip_runtime.h>
#include <hip/hip_bf16.h>

#define HID 256
#define NHE 20000

typedef float v2f __attribute__((ext_vector_type(2)));
typedef float v8f __attribute__((ext_vector_type(8)));

// ---------------- init: zero accumulators ----------------
__global__ __launch_bounds__(256) void zero_kernel(float* __restrict__ feats,
                                                   float* __restrict__ counts,
                                                   unsigned* __restrict__ maxbits) {
    long long i = (long long)blockIdx.x * 256 + threadIdx.x;
    if (i < (long long)NHE * HID) feats[i] = 0.0f;
    if (i < NHE) counts[i] = 0.0f;
    if (i == 0) *maxbits = 0u;
}

// ---------------- stage 1: scatter mean-pool nodes -> hyperedges ----------------
__global__ __launch_bounds__(256) void pool_scatter(const float* __restrict__ x,
                                                    const int* __restrict__ node_idx,
                                                    const int* __restrict__ hyper_idx,
                                                    float* __restrict__ feats,
                                                    float* __restrict__ counts) {
    int c = blockIdx.x;           // one connection per block
    int j = threadIdx.x;          // one column per thread
    int n = node_idx[c];
    int h = hyper_idx[c];
    atomicAdd(&feats[(long long)h * HID + j], x[(long long)n * HID + j]);
    if (j == 0) atomicAdd(&counts[h], 1.0f);
}

// feats /= clip(counts,1);  feats_upd = feats (normalized)
__global__ __launch_bounds__(256) void normalize_feats(float* __restrict__ feats,
                                                       const float* __restrict__ counts,
                                                       float* __restrict__ feats_upd) {
    int i = blockIdx.x;
    int j = threadIdx.x;
    float cnt = counts[i];
    cnt = cnt < 1.0f ? 1.0f : cnt;
    float v = feats[(long long)i * HID + j] / cnt;
    feats[(long long)i * HID + j] = v;
    feats_upd[(long long)i * HID + j] = v;
}

// ---------------- stage 2: fused per-hyperedge MLP via f32 WMMA ----------------
// M[e] = relu(feats[e] @ W1^T + b1) @ W2^T + b2   for all 20000 hyperedges.
// Block = 128 threads (4 waves), owns 16 rows. Each wave computes 4 16-col tiles.
__global__ __launch_bounds__(128) void mlp_wmma(const float* __restrict__ feats,
                                                const float* __restrict__ W1,
                                                const float* __restrict__ b1,
                                                const float* __restrict__ W2,
                                                const float* __restrict__ b2,
                                                float* __restrict__ Mout) {
    __shared__ float sA[16 * HID];   // input tile (16 rows x 256)
    __shared__ float sH[16 * HID];   // hidden tile after relu

    const int row0 = blockIdx.x * 16;
    const int tid  = threadIdx.x;

    // cooperative load: 16x256 floats = 1024 float4 by 128 threads
    {
        const float4* gsrc = (const float4*)(feats + (long long)row0 * HID);
        float4* ldst = (float4*)sA;
        #pragma unroll
        for (int i = 0; i < 8; ++i)
            ldst[tid + i * 128] = gsrc[tid + i * 128];
    }
    __syncthreads();

    const int wave = tid >> 5;        // 0..3
    const int lane = tid & 31;
    const int half = lane >> 4;       // 0/1
    const int r    = lane & 15;       // 0..15
    const int koff = 2 * half;        // A/B fragment K sub-offset

    // ---- GEMM1: H = relu(A @ W1^T + b1) ----
    for (int t = 0; t < 4; ++t) {
        const int col0 = (wave * 4 + t) * 16;
        v8f acc = {};
        #pragma unroll 4
        for (int k0 = 0; k0 < HID; k0 += 4) {
            // A frag: A[m=r][k0+koff .. +1]      (16x4 f32, 2 VGPRs)
            v2f a = *(const v2f*)&sA[r * HID + k0 + koff];
            // B frag: B[k][n] = W1[col0+n][k0+k] (4x16 f32, 2 VGPRs)
            v2f b = *(const v2f*)&W1[(long long)(col0 + r) * HID + k0 + koff];
            acc = __builtin_amdgcn_wmma_f32_16x16x4_f32(false, a, false, b,
                                                        (short)0, acc, false, false);
        }
        float bias = b1[col0 + r];
        #pragma unroll
        for (int v = 0; v < 8; ++v) {
            float h = acc[v] + bias;            // D element (m = v+8*half, n = col0+r)
            h = h > 0.0f ? h : 0.0f;
            sH[(v + 8 * half) * HID + col0 + r] = h;
        }
    }
    __syncthreads();

    // ---- GEMM2: M = H @ W2^T + b2 ----
    for (int t = 0; t < 4; ++t) {
        const int col0 = (wave * 4 + t) * 16;
        v8f acc = {};
        #pragma unroll 4
        for (int k0 = 0; k0 < HID; k0 += 4) {
            v2f a = *(const v2f*)&sH[r * HID + k0 + koff];
            v2f b = *(const v2f*)&W2[(long long)(col0 + r) * HID + k0 + koff];
            acc = __builtin_amdgcn_wmma_f32_16x16x4_f32(false, a, false, b,
                                                        (short)0, acc, false, false);
        }
        float bias = b2[col0 + r];
        #pragma unroll
        for (int v = 0; v < 8; ++v) {
            Mout[(long long)(row0 + v + 8 * half) * HID + col0 + r] = acc[v] + bias;
        }
    }
}

// ---------------- stage 3: dual-edge scatter: feats_upd[dst] += 0.1*M[src] ----------------
__global__ __launch_bounds__(256) void dual_scatter(const float* __restrict__ Mbuf,
                                                    const int* __restrict__ src,
                                                    const int* __restrict__ dst,
                                                    float* __restrict__ feats_upd) {
    int e = blockIdx.x;   // one dual edge per block
    int j = threadIdx.x;
    int s = src[e];
    int d = dst[e];
    atomicAdd(&feats_upd[(long long)d * HID + j], 0.1f * Mbuf[(long long)s * HID + j]);
}

// ---------------- stage 4: row L2 norm + global max ----------------
__global__ __launch_bounds__(256) void row_norm(const float* __restrict__ feats_upd,
                                                float* __restrict__ norms,
                                                unsigned* __restrict__ maxbits) {
    __shared__ float red[256];
    int i = blockIdx.x;
    int j = threadIdx.x;
    float v = feats_upd[(long long)i * HID + j];
    red[j] = v * v;
    __syncthreads();
    #pragma unroll
    for (int s = 128; s > 0; s >>= 1) {
        if (j < s) red[j] += red[j + s];
        __syncthreads();
    }
    if (j == 0) {
        float nrm = sqrtf(red[0]);
        norms[i] = nrm;
        // norms are non-negative: uint bit-pattern order == float order
        atomicMax(maxbits, __float_as_uint(nrm));
    }
}

__global__ __launch_bounds__(256) void finalize(const float* __restrict__ norms,
                                                const unsigned* __restrict__ maxbits,
                                                float* __restrict__ out) {
    int i = blockIdx.x * 256 + threadIdx.x;
    if (i >= NHE) return;
    float mx = __uint_as_float(*maxbits);
    out[i] = norms[i] / mx * 0.9f + 0.1f;
}

extern "C" void kernel_launch(void* const* d_in, const int* in_sizes, int n_in,
                              void* d_out, int out_size, void* d_ws, size_t ws_size,
                              hipStream_t stream) {
    // inputs (setup_inputs order):
    // 0: x           [100000, 256] f32
    // 1: edge_index  [2, 200000]   int32
    // 2: edge_weights[20000]       f32 (UNUSED by reference)
    // 3: dual_edges  [2, D]        int32
    // 4: W1 [256,256]  5: b1 [256]  6: W2 [256,256]  7: b2 [256]
    const float* x  = (const float*)d_in[0];
    const int*   ei = (const int*)d_in[1];
    const int*   de = (const int*)d_in[3];
    const float* W1 = (const float*)d_in[4];
    const float* b1 = (const float*)d_in[5];
    const float* W2 = (const float*)d_in[6];
    const float* b2 = (const float*)d_in[7];
    float* out = (float*)d_out;

    const int nconn  = in_sizes[1] / 2;
    const int nduals = in_sizes[3] / 2;
    const int* node_idx  = ei;
    const int* hyper_idx = ei + nconn;
    const int* dsrc = de;
    const int* ddst = de + nduals;

    // workspace layout (floats)
    float* ws        = (float*)d_ws;
    float* feats     = ws;                               // NHE*HID
    float* counts    = feats + (long long)NHE * HID;     // NHE
    float* Mbuf      = counts + NHE;                     // NHE*HID
    float* feats_upd = Mbuf + (long long)NHE * HID;      // NHE*HID
    float* norms     = feats_upd + (long long)NHE * HID; // NHE
    unsigned* maxbits = (unsigned*)(norms + NHE);        // 1

    // 0) zero accumulators (every call: determinism, no cross-call state)
    zero_kernel<<<(NHE * HID + 255) / 256, 256, 0, stream>>>(feats, counts, maxbits);

    // 1) mean-pool nodes into hyperedges
    pool_scatter<<<nconn, 256, 0, stream>>>(x, node_idx, hyper_idx, feats, counts);
    normalize_feats<<<NHE, 256, 0, stream>>>(feats, counts, feats_upd);

    // 2) per-hyperedge MLP message via f32 WMMA (16x16x4)
    mlp_wmma<<<NHE / 16, 128, 0, stream>>>(feats, W1, b1, W2, b2, Mbuf);

    // 3) dual-edge scatter-add
    if (nduals > 0)
        dual_scatter<<<nduals, 256, 0, stream>>>(Mbuf, dsrc, ddst, feats_upd);

    // 4) norms, max, final rescale
    row_norm<<<NHE, 256, 0, stream>>>(feats_upd, norms, maxbits);
    finalize<<<(NHE + 255) / 256, 256, 0, stream>>>(norms, maxbits, out);
}